// ReferenceLoss_19104014532666
// MI455X (gfx1250) — compile-verified
//
#include <hip/hip_runtime.h>
#include <hip/hip_bf16.h>

typedef __attribute__((ext_vector_type(2))) float v2f;
typedef __attribute__((ext_vector_type(8))) float v8f;

#define D 256
#define MARGIN 0.3f

// ---------------------------------------------------------------------------
// Kernel 1: per-row squared norms. One wave (32 lanes) per row.
// ---------------------------------------------------------------------------
__global__ __launch_bounds__(256) void sq_kernel(const float* __restrict__ X,
                                                 float* __restrict__ sq, int n) {
  int row  = blockIdx.x * 8 + (threadIdx.x >> 5);
  int lane = threadIdx.x & 31;
  if (row >= n) return;
  const float* x = X + (size_t)row * D;
  float s = 0.f;
  #pragma unroll
  for (int c = lane; c < D; c += 32) { float v = x[c]; s += v * v; }
  #pragma unroll
  for (int m = 16; m >= 1; m >>= 1) s += __shfl_xor(s, m);
  if (lane == 0) sq[row] = s;
}

// ---------------------------------------------------------------------------
// Kernel 2: fused Gram (fp32 WMMA 16x16x4) -> distance -> masked argmax/argmin.
// One block = one 16-row tile; 8 waves split the 512 column tiles.
// ---------------------------------------------------------------------------
__global__ __launch_bounds__(256) void gram_argminmax_kernel(
    const float* __restrict__ X, const int* __restrict__ tgt,
    const float* __restrict__ sq, int* __restrict__ c_pos,
    int* __restrict__ c_neg, int n) {
  __shared__ float ldsA[16 * D];          // 16 KB A slab, reused by all waves
  __shared__ float rvP[8][16]; __shared__ int riP[8][16];
  __shared__ float rvN[8][16]; __shared__ int riN[8][16];

  const int rowBase = blockIdx.x * 16;
  const int tid   = threadIdx.x;
  const int wave  = tid >> 5;
  const int lane  = tid & 31;
  const int lhalf = lane >> 4;            // half-wave selects K pair {0,1}/{2,3}
  const int l16   = lane & 15;            // A: row M ; B: column N (row of X)
  const int nTiles = n >> 4;

  // Cooperative, fully coalesced A-slab load (16 rows x 256 f32).
  {
    const float4* src = (const float4*)(X + (size_t)rowBase * D);
    float4* dst = (float4*)ldsA;
    #pragma unroll
    for (int t = tid; t < 16 * D / 4; t += 256) dst[t] = src[t];
  }
  __syncthreads();

  float sqRow[8]; int tgtRow[8];
  #pragma unroll
  for (int v = 0; v < 8; ++v) {
    int r = rowBase + v + 8 * lhalf;      // C/D layout: VGPR v -> M = v + 8*lhalf
    sqRow[v] = sq[r];
    tgtRow[v] = tgt[r];
  }

  float bestP[8], bestN[8]; int idxP[8], idxN[8];
  #pragma unroll
  for (int v = 0; v < 8; ++v) {
    bestP[v] = -INFINITY; idxP[v] = 0x7fffffff;
    bestN[v] =  INFINITY; idxN[v] = 0x7fffffff;
  }

  const float* aptr = ldsA + l16 * D + 2 * lhalf;
  for (int jt = wave; jt < nTiles; jt += 8) {
    const int colBase = jt * 16;
    const int col = colBase + l16;        // this lane's column N
    const float* bptr = X + (size_t)col * D + 2 * lhalf;

    v8f acc = {};
    #pragma unroll
    for (int k = 0; k < D; k += 4) {
      v2f a = *(const v2f*)(aptr + k);    // ds_load_b64 from LDS
      v2f b = *(const v2f*)(bptr + k);    // global_load_b64, L2-resident
      acc = __builtin_amdgcn_wmma_f32_16x16x4_f32(
          false, a, false, b, (short)0, acc, false, false);
    }

    float sqC = sq[col];
    int   tgC = tgt[col];
    #pragma unroll
    for (int v = 0; v < 8; ++v) {
      float d2 = sqRow[v] + sqC - 2.0f * acc[v];
      float dd = __builtin_sqrtf(fmaxf(d2, 1e-12f));
      if (tgtRow[v] == tgC) {
        if (dd > bestP[v]) { bestP[v] = dd; idxP[v] = col; }   // strict > keeps
      } else {                                                 // first occurrence
        if (dd < bestN[v]) { bestN[v] = dd; idxN[v] = col; }
      }
    }
  }

  // Reduce across the 16 lanes of each half-wave (xor masks 1..8 stay in-half),
  // tie-break to the lower column index (jnp first-occurrence semantics).
  #pragma unroll
  for (int v = 0; v < 8; ++v) {
    float pv = bestP[v]; int pi = idxP[v];
    float nv = bestN[v]; int ni = idxN[v];
    #pragma unroll
    for (int m = 1; m < 16; m <<= 1) {
      float opv = __shfl_xor(pv, m); int opi = __shfl_xor(pi, m);
      float onv = __shfl_xor(nv, m); int oni = __shfl_xor(ni, m);
      if (opv > pv || (opv == pv && opi < pi)) { pv = opv; pi = opi; }
      if (onv < nv || (onv == nv && oni < ni)) { nv = onv; ni = oni; }
    }
    if (l16 == 0) {                       // lane 0 -> row v, lane 16 -> row v+8
      int r = v + 8 * lhalf;
      rvP[wave][r] = pv; riP[wave][r] = pi;
      rvN[wave][r] = nv; riN[wave][r] = ni;
    }
  }
  __syncthreads();

  if (tid < 16) {                         // combine the 8 waves, same tie-break
    float pv = rvP[0][tid]; int pi = riP[0][tid];
    float nv = rvN[0][tid]; int ni = riN[0][tid];
    #pragma unroll
    for (int w = 1; w < 8; ++w) {
      float opv = rvP[w][tid]; int opi = riP[w][tid];
      float onv = rvN[w][tid]; int oni = riN[w][tid];
      if (opv > pv || (opv == pv && opi < pi)) { pv = opv; pi = opi; }
      if (onv < nv || (onv == nv && oni < ni)) { nv = onv; ni = oni; }
    }
    c_pos[rowBase + tid] = pi;
    c_neg[rowBase + tid] = ni;
  }
}

// ---------------------------------------------------------------------------
// Kernel 3: ranks j,k within masked subsets + the three distances + per-row
// loss term. One wave per row.
// ---------------------------------------------------------------------------
__global__ __launch_bounds__(256) void rank_loss_kernel(
    const float* __restrict__ X, const int* __restrict__ tgt,
    const float* __restrict__ sq, const int* __restrict__ c_pos,
    const int* __restrict__ c_neg, float* __restrict__ loss, int n) {
  int row  = blockIdx.x * 8 + (threadIdx.x >> 5);
  int lane = threadIdx.x & 31;
  if (row >= n) return;

  int ti = tgt[row];
  int cp = c_pos[row];
  int cn = c_neg[row];

  int cntP = 0, cntN = 0;
  for (int l = lane; l < n; l += 32) {    // rank = #mask-true entries at cols<=c
    bool same = (tgt[l] == ti);
    cntP += (same  && l <= cp);
    cntN += (!same && l <= cn);
  }
  #pragma unroll
  for (int m = 16; m >= 1; m >>= 1) {
    cntP += __shfl_xor(cntP, m);
    cntN += __shfl_xor(cntN, m);
  }
  int j = cntP - 1; if (j < 0) j = 0;     // replicated indexing "bug": rank used
  int k = cntN - 1; if (k < 0) k = 0;     // as a raw column index into dist

  const float* xi = X + (size_t)row * D;
  const float* xj = X + (size_t)j * D;
  const float* xk = X + (size_t)k * D;
  float sij = 0.f, sik = 0.f, sjk = 0.f;
  #pragma unroll
  for (int c = lane; c < D; c += 32) {
    float a = xi[c], b = xj[c], d = xk[c];
    sij += a * b; sik += a * d; sjk += b * d;
  }
  #pragma unroll
  for (int m = 16; m >= 1; m >>= 1) {
    sij += __shfl_xor(sij, m);
    sik += __shfl_xor(sik, m);
    sjk += __shfl_xor(sjk, m);
  }
  if (lane == 0) {
    float dij = __builtin_sqrtf(fmaxf(sq[row] + sq[j] - 2.f * sij, 1e-12f));
    float dik = __builtin_sqrtf(fmaxf(sq[row] + sq[k] - 2.f * sik, 1e-12f));
    float djk = __builtin_sqrtf(fmaxf(sq[j]   + sq[k] - 2.f * sjk, 1e-12f));
    loss[row] = fmaxf(dij - dik + MARGIN, 0.f) + fabsf(dik - djk);
  }
}

// ---------------------------------------------------------------------------
// Kernel 4: deterministic single-block tree reduction -> scalar mean.
// ---------------------------------------------------------------------------
__global__ __launch_bounds__(256) void reduce_kernel(const float* __restrict__ loss,
                                                     float* __restrict__ out, int n) {
  __shared__ float buf[256];
  int tid = threadIdx.x;
  float s = 0.f;
  for (int i = tid; i < n; i += 256) s += loss[i];   // fixed order: deterministic
  buf[tid] = s;
  __syncthreads();
  #pragma unroll
  for (int stride = 128; stride > 0; stride >>= 1) {
    if (tid < stride) buf[tid] += buf[tid + stride];
    __syncthreads();
  }
  if (tid == 0) out[0] = buf[0] * (1.0f / (float)n);
}

// ---------------------------------------------------------------------------
extern "C" void kernel_launch(void* const* d_in, const int* in_sizes, int n_in,
                              void* d_out, int out_size, void* d_ws, size_t ws_size,
                              hipStream_t stream) {
  const float* X   = (const float*)d_in[0];
  const int*   tgt = (const int*)d_in[1];
  const int n = in_sizes[1];              // 8192

  char* ws = (char*)d_ws;
  float* sq    = (float*)(ws);                       //  n f32
  int*   c_pos = (int*)(ws + (size_t)n * 4);         //  n i32
  int*   c_neg = (int*)(ws + (size_t)n * 8);         //  n i32
  float* loss  = (float*)(ws + (size_t)n * 12);      //  n f32

  sq_kernel<<<(n + 7) / 8, 256, 0, stream>>>(X, sq, n);
  gram_argminmax_kernel<<<n / 16, 256, 0, stream>>>(X, tgt, sq, c_pos, c_neg, n);
  rank_loss_kernel<<<(n + 7) / 8, 256, 0, stream>>>(X, tgt, sq, c_pos, c_neg, loss, n);
  reduce_kernel<<<1, 256, 0, stream>>>(loss, (float*)d_out, n);
}